// Quantize_32384053411991
// MI455X (gfx1250) — compile-verified
//
#include <hip/hip_runtime.h>

typedef float v2f __attribute__((ext_vector_type(2)));
typedef float v8f __attribute__((ext_vector_type(8)));

// D = A(16x4) * B(4x16) + C ; fp32 WMMA (CDNA5), 8-arg form:
// (neg_a, A, neg_b, B, c_mod, C, reuse_a, reuse_b)
#define WMMA_F32_16x16x4(A, B, C) \
  __builtin_amdgcn_wmma_f32_16x16x4_f32(false, (A), false, (B), (short)0, (C), false, false)

// Problem constants (from reference shapes)
#define NB   16        // batch
#define CH   256       // C (= D)
#define HW   1024      // H*W
#define NTOK 16384     // NB*HW
#define KCB  8192      // codebook entries
#define QELT 4194304   // NB*CH*HW  (quantized output elements)
#define DIFF_OFF  4194304
#define IDX_OFF   4194305

#define ROWS 64        // tokens per workgroup (M-blocking for codebook reuse)

// -------- Kernel 1: codebook column norms + zero diff accumulator --------
__global__ __launch_bounds__(256) void vq_cbnorm_kernel(
    const float* __restrict__ cb, float* __restrict__ cbnorm,
    float* __restrict__ out) {
  int k = blockIdx.x * 256 + threadIdx.x;   // 8192 threads total
  float s = 0.0f;
  for (int d = 0; d < CH; ++d) {            // coalesced over k (stride-1)
    float v = cb[(size_t)d * KCB + k];
    s += v * v;
  }
  cbnorm[k] = s;
  if (k == 0) out[DIFF_OFF] = 0.0f;         // reset loss accumulator each call
}

// -------- Kernel 2: fused conv (WMMA) + distance GEMM (WMMA) + argmin + loss --
// 8 waves = 4 row-groups x 2 k-splits. The 4 waves of one k-split walk an
// identical codebook-address sequence, so WGP$ filters B redundancy (4x less
// L2 traffic than one-row-tile-per-block scheduling).
__global__ __launch_bounds__(256) void vq_fused_kernel(
    const float* __restrict__ x, const float* __restrict__ conv_w,
    const float* __restrict__ conv_b, const float* __restrict__ cb,
    const float* __restrict__ cbnorm, int* __restrict__ idx_ws,
    float* __restrict__ out) {
  __shared__ float xs[ROWS * 258];   // xp tile (64x256), padded rows
  __shared__ float rs[8 * 8 * 32];   // per-wave argmin candidates: score
  __shared__ int   ri[8 * 8 * 32];   // per-wave argmin candidates: index
  __shared__ int   ridxfin[ROWS];    // final index per row

  const int tid   = threadIdx.x;
  const int wv    = tid >> 5;        // wave id 0..7
  const int lane  = tid & 31;
  const int col   = lane & 15;       // A row / B,C col within the 16-tile
  const int khalf = lane >> 4;       // 0 or 1 (K-halves in WMMA layout)
  const int g     = wv & 3;          // row-group: rows g*16 .. g*16+15
  const int s     = wv >> 2;         // k-split (0/1) and d-half in phase 1
  const int n0    = blockIdx.x * ROWS;

  v2f afrag[64];                     // 128 VGPRs: full K=256 A fragment

  // ---- Phase 1: 1x1 conv -> xp tile (64 x 256) via WMMA, staged in LDS ----
  {
    int n    = n0 + g * 16 + col;
    int bimg = n >> 10;
    int hw   = n & 1023;
    const float* xb = x + (size_t)bimg * (CH * HW) + hw;
#pragma unroll
    for (int p = 0; p < 64; ++p) {   // A[c] = x[b, c, hw]; K = 4p + 2*khalf + j
      int c = 4 * p + 2 * khalf;
      afrag[p].x = xb[(size_t)c * HW];
      afrag[p].y = xb[(size_t)(c + 1) * HW];
    }
    for (int u = 0; u < 8; ++u) {    // this wave covers d-half s (8 d-tiles)
      int d0 = s * 128 + u * 16;
      v8f acc = {0.f, 0.f, 0.f, 0.f, 0.f, 0.f, 0.f, 0.f};
      const float* wb = conv_w + (size_t)(d0 + col) * CH + 2 * khalf;
#pragma unroll
      for (int p = 0; p < 64; ++p) { // B[kc][d] = conv_w[d][c+kc], contiguous pair
        v2f bm = *(const v2f*)(wb + 4 * p);
        acc = WMMA_F32_16x16x4(afrag[p], bm, acc);
      }
      float bias = conv_b[d0 + col];
#pragma unroll
      for (int i = 0; i < 8; ++i)    // C layout: row = i + 8*khalf, col = lane&15
        xs[(g * 16 + i + 8 * khalf) * 258 + d0 + col] = acc[i] + bias;
    }
  }
  __syncthreads();

  // ---- Phase 2: reload A fragment = this row-group's xp rows (K over d) ----
#pragma unroll
  for (int p = 0; p < 64; ++p)
    afrag[p] = *(const v2f*)&xs[(g * 16 + col) * 258 + 4 * p + 2 * khalf];

  // ---- Phase 3: scores S = xp . cb ; track argmax(2S - ||cb||^2) ----
  float best[8];
  int   bidx[8];
#pragma unroll
  for (int i = 0; i < 8; ++i) { best[i] = -3.4e38f; bidx[i] = 0; }

#pragma unroll 1
  for (int t = 0; t < 256; ++t) {    // k-split s covers k0 = (2t+s)*16
    int k0 = (2 * t + s) * 16;
    const float* cbb = cb + (size_t)(2 * khalf) * KCB + k0 + col;
    v8f acc = {0.f, 0.f, 0.f, 0.f, 0.f, 0.f, 0.f, 0.f};
#pragma unroll
    for (int p = 0; p < 64; ++p) {   // B[kc][k] = cb[d][k], d = 4p + 2*khalf + j
      v2f bm;
      bm.x = cbb[(size_t)(4 * p) * KCB];
      bm.y = cbb[(size_t)(4 * p + 1) * KCB];
      acc = WMMA_F32_16x16x4(afrag[p], bm, acc);
    }
    float cbn = cbnorm[k0 + col];
    int   myk = k0 + col;
#pragma unroll
    for (int i = 0; i < 8; ++i) {    // -dist = 2S - ||cb||^2 (+||x||^2 const)
      float v = 2.0f * acc[i] - cbn;
      if (v > best[i]) { best[i] = v; bidx[i] = myk; } // k increases per lane
    }
    if ((t & 7) == 7) __syncthreads(); // keep co-walking waves in WGP$ window
  }

  // cross-lane / cross-k-split argmin reduction via LDS
#pragma unroll
  for (int i = 0; i < 8; ++i) {
    rs[(wv * 8 + i) * 32 + lane] = best[i];
    ri[(wv * 8 + i) * 32 + lane] = bidx[i];
  }
  __syncthreads();
  if (tid < ROWS) {
    int r = tid;                      // row within block
    int gg = r >> 4, rr = r & 15, ii = rr & 7, half = rr >> 3;
    float bs = -3.4e38f;
    int   bk = 0x7fffffff;
    for (int ss = 0; ss < 2; ++ss) {  // two k-split waves own this row
      int w = ss * 4 + gg;
      for (int l = 0; l < 16; ++l) {
        float v = rs[(w * 8 + ii) * 32 + half * 16 + l];
        int   k = ri[(w * 8 + ii) * 32 + half * 16 + l];
        if (v > bs || (v == bs && k < bk)) { bs = v; bk = k; } // first-match tie
      }
    }
    ridxfin[r] = bk;
    idx_ws[n0 + r] = bk;
    out[IDX_OFF + n0 + r] = (float)bk;
  }
  __syncthreads();

  // ---- Phase 4: commitment loss partial from live LDS xp tile ----
  {
    int r = tid >> 2, j = tid & 3;    // 4 threads per row, 64 d each
    int kb = ridxfin[r];
    float sum = 0.0f;
#pragma unroll
    for (int m = 0; m < 64; ++m) {
      int d = j + 4 * m;
      float e = cb[(size_t)d * KCB + kb] - xs[r * 258 + d];
      sum += e * e;
    }
    rs[tid] = sum;                    // reuse reduction LDS
    __syncthreads();
    if (tid == 0) {
      float tot = 0.0f;
      for (int q = 0; q < 256; ++q) tot += rs[q];
      atomicAdd(out + DIFF_OFF, tot * (1.0f / (float)QELT));
    }
  }
}

// -------- Kernel 3: gather quantized codes, (b,d,hw)-major coalesced writes ---
__global__ __launch_bounds__(256) void vq_gather_kernel(
    const float* __restrict__ cb, const int* __restrict__ idx_ws,
    float* __restrict__ out) {
  int tid = blockIdx.x * 256 + threadIdx.x;  // QELT threads
  int hw = tid & 1023;
  int d  = (tid >> 10) & 255;
  int b  = tid >> 18;
  int k  = idx_ws[b * HW + hw];              // coalesced
  out[tid] = cb[(size_t)d * KCB + k];        // L2-resident gather
}

extern "C" void kernel_launch(void* const* d_in, const int* in_sizes, int n_in,
                              void* d_out, int out_size, void* d_ws, size_t ws_size,
                              hipStream_t stream) {
  const float* x      = (const float*)d_in[0];   // (16,256,32,32)
  const float* conv_w = (const float*)d_in[1];   // (256,256)
  const float* conv_b = (const float*)d_in[2];   // (256,)
  const float* cb     = (const float*)d_in[3];   // (256,8192)
  float* out = (float*)d_out;

  float* cbnorm = (float*)d_ws;                  // 8192 floats
  int*   idx_ws = (int*)((char*)d_ws + KCB * sizeof(float)); // 16384 ints

  vq_cbnorm_kernel<<<KCB / 256, 256, 0, stream>>>(cb, cbnorm, out);
  vq_fused_kernel<<<NTOK / ROWS, 256, 0, stream>>>(x, conv_w, conv_b, cb,
                                                   cbnorm, idx_ws, out);
  vq_gather_kernel<<<QELT / 256, 256, 0, stream>>>(cb, idx_ws, out);
}